// Kernel_Correlation_61667140436412
// MI455X (gfx1250) — compile-verified
//
#include <hip/hip_runtime.h>

// Kernel correlation for MI455X (gfx1250, wave32, WMMA).
//
// d2 = |g|^2 + |k|^2 - 2 g.k. We compute  -log2(e)*d2  entirely inside one
// V_WMMA_F32_16X16X4_F32 per (4 points x 1 kernel):
//   A (16x4)  row = gathered point vector [gx, gy, gz, 1]
//   B (4x16)  col = kernel point, rows [2L*kx, 2L*ky, 2L*kz, -L*|k|^2], L=log2(e)
//   C_init    C[M][N] = -L*|g_M|^2   (constant across N == exactly C's layout)
//   D = A*B + C_init = -L*d2  ->  exp2(D) = exp(-d2), one v_exp_f32 per element.
// Column reduction uses DPP row_ror butterflies (VALU) instead of ds_bpermute,
// keeping the DS pipe nearly idle so exp (TRANS) + WMMA co-execute freely.

typedef __attribute__((ext_vector_type(2))) float v2f;
typedef __attribute__((ext_vector_type(8))) float v8f;

#define N_KER 64
#define K_PTS 16
#define WAVES_PER_BLOCK 8
#define PTS_PER_WAVE 4
#define PTS_PER_BLOCK (WAVES_PER_BLOCK * PTS_PER_WAVE)   // 32 points / block

template <int CTRL>
__device__ __forceinline__ float dpp_ror_add(float x) {
  int y = __builtin_amdgcn_update_dpp(0, __builtin_bit_cast(int, x),
                                      CTRL, 0xF, 0xF, true);
  return x + __builtin_bit_cast(float, y);
}

// Sum across the 16 lanes of each half-wave (DPP row == 16 lanes on wave32).
__device__ __forceinline__ float rowsum16(float x) {
  x = dpp_ror_add<0x128>(x);   // row_ror:8
  x = dpp_ror_add<0x124>(x);   // row_ror:4
  x = dpp_ror_add<0x122>(x);   // row_ror:2
  x = dpp_ror_add<0x121>(x);   // row_ror:1
  return x;
}

__global__ __launch_bounds__(256) void kernel_corr_wmma(
    const float* __restrict__ normal,     // [n*3]
    const int*   __restrict__ neighbour,  // [n*3]
    const float* __restrict__ kern,       // [64,16,3]
    float*       __restrict__ out,        // [n*64]
    int n)
{
  // Transformed B operand per (kernel m, lane): 64*32*8B = 16 KB
  __shared__ v2f sB[N_KER * 32];
  // Per-wave {slo,shi} pairs: [wave][half][m] float2 = 8 KB
  __shared__ v2f sPair[WAVES_PER_BLOCK * 2 * N_KER];

  const int tid  = threadIdx.x;
  const int lane = tid & 31;
  const int wave = tid >> 5;

  const float L = 1.4426950408889634f;   // log2(e)

  // ---- Stage B into LDS (once per block). 32-bit B 4x16 layout assumed as
  // mirror of the documented A 16x4 layout:
  //   VGPR0: lanes 0-15 K=0 (col=lane), lanes 16-31 K=2 (col=lane-16)
  //   VGPR1: lanes 0-15 K=1,            lanes 16-31 K=3
  for (int e = tid; e < N_KER * 32; e += blockDim.x) {
    const int m   = e >> 5;
    const int l   = e & 31;
    const int col = l & 15;
    const float* kp = kern + (m * K_PTS + col) * 3;
    const float kx = kp[0], ky = kp[1], kz = kp[2];
    v2f b;
    if (l < 16) { b.x = 2.0f * L * kx; b.y = 2.0f * L * ky; }
    else        { b.x = 2.0f * L * kz; b.y = -L * (kx*kx + ky*ky + kz*kz); }
    sB[e] = b;
  }
  __syncthreads();

  // ---- Each wave owns 4 points; 16 A-rows: row = q*4 + p
  // (q = point-in-wave, p = gather slot: 0=self, 1..3=neighbours)
  const int pt_base = blockIdx.x * PTS_PER_BLOCK + wave * PTS_PER_WAVE;
  const int row = lane & 15;
  const int q   = row >> 2;
  const int p   = row & 3;
  int point = pt_base + q;
  if (point >= n) point = n - 1;                       // clamp for tail block
  const int gidx = (p == 0) ? point : neighbour[point * 3 + (p - 1)];
  const float* gp = normal + (long long)gidx * 3;

  // A 16x4 f32 layout: lanes 0-15 -> (K=0,K=1) = (gx,gy); lanes 16-31 -> (gz, 1)
  v2f a;
  const float gz = gp[2];
  if (lane < 16) { a.x = gp[0]; a.y = gp[1]; }
  else           { a.x = gz;    a.y = 1.0f;  }

  // |g|^2 per row (valid on lanes 0..15): pull gz from partner lane.
  const float zOther = __shfl_xor(a.x, 16, 32);
  const float g2 = a.x * a.x + a.y * a.y + zOther * zOther;

  // C layout: VGPR v holds row v (lanes 0-15) and row v+8 (lanes 16-31).
  // Seed C with -L*|g|^2 so the WMMA emits -L*d2 directly.
  v8f cinit;
  const int hb = (lane >= 16) ? 8 : 0;
  #pragma unroll
  for (int v = 0; v < 8; ++v) cinit[v] = -L * __shfl(g2, v + hb, 32);

  const int half = lane >> 4;           // 0: q0/q1 sums, 1: q2/q3 sums
  v2f* myPair = sPair + (wave * 2 + half) * N_KER;
  const bool writer = (lane & 15) == 0; // lanes 0 and 16

  #pragma unroll 2
  for (int m = 0; m < N_KER; ++m) {
    const v2f b = sB[m * 32 + lane];
    // v_wmma_f32_16x16x4_f32 : D = A(16x4) x B(4x16) + C = -L*d2
    v8f d = __builtin_amdgcn_wmma_f32_16x16x4_f32(
        /*neg_a=*/false, a, /*neg_b=*/false, b,
        /*c_mod=*/(short)0, cinit, /*reuse_a=*/false, /*reuse_b=*/false);

    // exp(-d2) = exp2(-L*d2): one v_exp_f32 each (TRANS pipe)
    const float e0 = __builtin_amdgcn_exp2f(d[0]);
    const float e1 = __builtin_amdgcn_exp2f(d[1]);
    const float e2 = __builtin_amdgcn_exp2f(d[2]);
    const float e3 = __builtin_amdgcn_exp2f(d[3]);
    const float e4 = __builtin_amdgcn_exp2f(d[4]);
    const float e5 = __builtin_amdgcn_exp2f(d[5]);
    const float e6 = __builtin_amdgcn_exp2f(d[6]);
    const float e7 = __builtin_amdgcn_exp2f(d[7]);

    // Rows of one point are summed anyway -> pre-add the 4 C-VGPRs, then a
    // single 16-lane DPP reduction per point-pair.
    float slo = (e0 + e1) + (e2 + e3);  // rows 0-3 (q0) / 8-11 (q2)
    float shi = (e4 + e5) + (e6 + e7);  // rows 4-7 (q1) / 12-15 (q3)
    slo = rowsum16(slo);
    shi = rowsum16(shi);

    if (writer) {
      v2f r; r.x = slo; r.y = shi;      // {q(2h), q(2h+1)} for kernel m
      myPair[m] = r;                    // one ds_store_b64
    }
  }
  __syncthreads();

  // ---- Coalesced block store: contiguous 2048 floats of out.
  // pt_off = wave*4 + half*2 + d ; value at sPair[(wave*2+half)*64 + m][d]
  const float scale = 1.0f / 128.0f;    // 1/(2*sigma^2) * 1/(K_PTS*4)
  const float* sflat = (const float*)sPair;
  const long long obase = (long long)blockIdx.x * PTS_PER_BLOCK * N_KER;
  for (int e = tid; e < PTS_PER_BLOCK * N_KER; e += blockDim.x) {
    const int pt_off = e >> 6;          // 0..31
    const int m  = e & 63;
    const int w  = pt_off >> 2;
    const int h  = (pt_off >> 1) & 1;
    const int dd = pt_off & 1;
    if (blockIdx.x * PTS_PER_BLOCK + pt_off < n)
      out[obase + e] = sflat[(((w * 2 + h) * N_KER) + m) * 2 + dd] * scale;
  }
}

extern "C" void kernel_launch(void* const* d_in, const int* in_sizes, int n_in,
                              void* d_out, int out_size, void* d_ws, size_t ws_size,
                              hipStream_t stream) {
  const float* normal    = (const float*)d_in[0];
  const int*   neighbour = (const int*)d_in[1];
  const float* kern      = (const float*)d_in[2];
  float* out = (float*)d_out;
  const int n = in_sizes[0] / 3;
  const int blocks = (n + PTS_PER_BLOCK - 1) / PTS_PER_BLOCK;
  kernel_corr_wmma<<<blocks, 256, 0, stream>>>(normal, neighbour, kern, out, n);
  (void)d_ws; (void)ws_size; (void)out_size; (void)n_in;
}